// GCN_6064493822473
// MI455X (gfx1250) — compile-verified
//
#include <hip/hip_runtime.h>
#include <hip/hip_bf16.h>

// ---------------------------------------------------------------------------
// GCN forward (2x GCNConv, D=64) for MI455X / gfx1250, wave32.
// Dense GEMMs use V_WMMA_F32_16X16X4_F32 (exact fp32 path).
// Edge aggregation uses global float atomics (memory-bound part).
// ---------------------------------------------------------------------------

typedef __attribute__((ext_vector_type(2))) float v2f;
typedef __attribute__((ext_vector_type(8))) float v8f;

#define D 64  // D_IN == D_HID == 64

// ---------------- degree / normalization ----------------

__global__ void deg_init(float* __restrict__ deg, int N) {
  int i = blockIdx.x * blockDim.x + threadIdx.x;
  if (i < N) deg[i] = 1.0f;  // self-loop
}

__global__ void deg_count(const int* __restrict__ dst, float* __restrict__ deg, int E) {
  int e = blockIdx.x * blockDim.x + threadIdx.x;
  if (e < E) atomicAdd(&deg[dst[e]], 1.0f);
}

__global__ void deg_rsqrt(float* __restrict__ deg, int N) {
  int i = blockIdx.x * blockDim.x + threadIdx.x;
  if (i < N) deg[i] = rsqrtf(deg[i]);
}

// ---------------- dense GEMM: Y[N,64] = f(X)[N,64] @ W[64,64] ----------------
// One wave computes a 16-row x 64-col tile with 4 f32 WMMA accumulators.
// RELU_BIAS fuses X' = relu(X + bias) into the A-operand load (layer-2 input).

template <bool RELU_BIAS>
__global__ void gemm64_wmma(const float* __restrict__ X, const float* __restrict__ W,
                            const float* __restrict__ bias, float* __restrict__ Y,
                            int nTiles) {
  __shared__ float Wlds[D * D];  // 16 KB
  // cooperative stage of W into LDS (float4)
  for (int i = threadIdx.x; i < (D * D) / 4; i += blockDim.x) {
    ((float4*)Wlds)[i] = ((const float4*)W)[i];
  }
  __syncthreads();

  const int wave = threadIdx.x >> 5;
  const int lane = threadIdx.x & 31;
  const int tile = blockIdx.x * (blockDim.x >> 5) + wave;
  if (tile >= nTiles) return;  // wave-uniform: EXEC stays all-ones below

  const int lrow  = lane & 15;       // row within tile for A; col within n-tile for B/C
  const int khalf = lane >> 4;       // 0: K pair {0,1}, 1: K pair {2,3}
  const float* xrow = X + (size_t)(tile * 16 + lrow) * D;

  v8f acc0 = {}, acc1 = {}, acc2 = {}, acc3 = {};

  for (int k0 = 0; k0 < D; k0 += 4) {
    const int kb = k0 + khalf * 2;
    // A fragment: 16x4 f32 -> 2 VGPRs per lane
    float a0 = xrow[kb + 0];
    float a1 = xrow[kb + 1];
    if (RELU_BIAS) {
      a0 = fmaxf(a0 + bias[kb + 0], 0.0f);
      a1 = fmaxf(a1 + bias[kb + 1], 0.0f);
    }
    v2f a;
    a.x = a0;
    a.y = a1;
    // B fragments: 4x16 f32 per n-tile -> 2 VGPRs per lane, from LDS
#pragma unroll
    for (int nt = 0; nt < 4; ++nt) {
      const int n = nt * 16 + lrow;
      v2f b;
      b.x = Wlds[(kb + 0) * D + n];
      b.y = Wlds[(kb + 1) * D + n];
      switch (nt) {
        case 0: acc0 = __builtin_amdgcn_wmma_f32_16x16x4_f32(false, a, false, b, (short)0, acc0, false, false); break;
        case 1: acc1 = __builtin_amdgcn_wmma_f32_16x16x4_f32(false, a, false, b, (short)0, acc1, false, false); break;
        case 2: acc2 = __builtin_amdgcn_wmma_f32_16x16x4_f32(false, a, false, b, (short)0, acc2, false, false); break;
        case 3: acc3 = __builtin_amdgcn_wmma_f32_16x16x4_f32(false, a, false, b, (short)0, acc3, false, false); break;
      }
    }
  }

  // C/D layout: VGPR r, lanes 0-15: M=r, N=lane; lanes 16-31: M=8+r, N=lane-16
  const int mbase = tile * 16 + khalf * 8;
#pragma unroll
  for (int r = 0; r < 8; ++r) {
    float* yrow = Y + (size_t)(mbase + r) * D;
    yrow[0 * 16 + lrow] = acc0[r];
    yrow[1 * 16 + lrow] = acc1[r];
    yrow[2 * 16 + lrow] = acc2[r];
    yrow[3 * 16 + lrow] = acc3[r];
  }
}

// scalar tail for rows not covered by 16-row tiles (unused when N%16==0)
template <bool RELU_BIAS>
__global__ void gemm64_tail(const float* __restrict__ X, const float* __restrict__ W,
                            const float* __restrict__ bias, float* __restrict__ Y,
                            int row0, int N) {
  int t = blockIdx.x * blockDim.x + threadIdx.x;
  int row = row0 + t / D;
  int col = t % D;
  if (row >= N) return;
  float s = 0.0f;
  for (int k = 0; k < D; ++k) {
    float x = X[(size_t)row * D + k];
    if (RELU_BIAS) x = fmaxf(x + bias[k], 0.0f);
    s += x * W[k * D + col];
  }
  Y[(size_t)row * D + col] = s;
}

// ---------------- aggregation ----------------

// agg[i,:] = dinv[i]^2 * xw[i,:]  (+ bias)   -- self-loop term (+ final bias)
__global__ void self_msg(const float* __restrict__ xw, const float* __restrict__ dinv,
                         const float* __restrict__ bias, float* __restrict__ agg,
                         int N, int addBias) {
  int t = blockIdx.x * blockDim.x + threadIdx.x;
  if (t >= N * (D / 4)) return;
  int i = t / (D / 4);
  int d = (t % (D / 4)) * 4;
  float nrm = dinv[i];
  nrm *= nrm;
  float4 v = ((const float4*)(xw + (size_t)i * D + d))[0];
  v.x *= nrm; v.y *= nrm; v.z *= nrm; v.w *= nrm;
  if (addBias) {
    v.x += bias[d + 0]; v.y += bias[d + 1]; v.z += bias[d + 2]; v.w += bias[d + 3];
  }
  ((float4*)(agg + (size_t)i * D + d))[0] = v;
}

// per edge: agg[dst,:] += xw[src,:] * dinv[src] * dinv[dst]
// 16 threads per edge, 4 dims each (float4 load, 4 scalar atomics)
__global__ void scatter_edges(const int* __restrict__ src, const int* __restrict__ dst,
                              const float* __restrict__ dinv, const float* __restrict__ xw,
                              float* __restrict__ agg, int E) {
  long long t = (long long)blockIdx.x * blockDim.x + threadIdx.x;
  long long e = t >> 4;
  if (e >= E) return;
  int d = (int)(t & 15) * 4;
  int s = src[e];
  int q = dst[e];
  float nrm = dinv[s] * dinv[q];
  float4 v = ((const float4*)(xw + (size_t)s * D + d))[0];
  float* out = agg + (size_t)q * D + d;
  atomicAdd(out + 0, v.x * nrm);
  atomicAdd(out + 1, v.y * nrm);
  atomicAdd(out + 2, v.z * nrm);
  atomicAdd(out + 3, v.w * nrm);
}

// ---------------------------------------------------------------------------

extern "C" void kernel_launch(void* const* d_in, const int* in_sizes, int n_in,
                              void* d_out, int out_size, void* d_ws, size_t ws_size,
                              hipStream_t stream) {
  const float* ndata = (const float*)d_in[0];
  const float* W0    = (const float*)d_in[1];
  const float* b0    = (const float*)d_in[2];
  const float* W1    = (const float*)d_in[3];
  const float* b1    = (const float*)d_in[4];
  const int* edge_src = (const int*)d_in[5];
  const int* edge_dst = (const int*)d_in[6];

  const int N = in_sizes[0] / D;   // 100000
  const int E = in_sizes[5];       // 1600000

  // workspace layout (16B-aligned offsets)
  size_t Npad = ((size_t)N + 3) & ~(size_t)3;
  float* dinv = (float*)d_ws;                 // N floats
  float* xw   = dinv + Npad;                  // N*64 floats
  float* agg  = xw + (size_t)N * D;           // N*64 floats
  float* out  = (float*)d_out;                // N*64 floats

  const int T = 256;

  // 1) degree (incl. self-loop) -> dinv
  deg_init<<<(N + T - 1) / T, T, 0, stream>>>(dinv, N);
  deg_count<<<(E + T - 1) / T, T, 0, stream>>>(edge_dst, dinv, E);
  deg_rsqrt<<<(N + T - 1) / T, T, 0, stream>>>(dinv, N);

  const int nTiles = N / 16;
  const int remRow0 = nTiles * 16;
  const int wavesPerBlk = T / 32;
  const int gemmGrid = (nTiles + wavesPerBlk - 1) / wavesPerBlk;

  // 2) layer 0: xw = ndata @ W0
  gemm64_wmma<false><<<gemmGrid, T, 0, stream>>>(ndata, W0, nullptr, xw, nTiles);
  if (remRow0 < N)
    gemm64_tail<false><<<((N - remRow0) * D + T - 1) / T, T, 0, stream>>>(ndata, W0, nullptr, xw, remRow0, N);

  // 3) agg = dinv^2 * xw (self-loop), then scatter edges
  self_msg<<<(N * (D / 4) + T - 1) / T, T, 0, stream>>>(xw, dinv, nullptr, agg, N, 0);
  {
    long long threads = (long long)E * 16;
    int grid = (int)((threads + T - 1) / T);
    scatter_edges<<<grid, T, 0, stream>>>(edge_src, edge_dst, dinv, xw, agg, E);
  }

  // 4) layer 1: xw = relu(agg + b0) @ W1   (bias+relu fused into A-load)
  gemm64_wmma<true><<<gemmGrid, T, 0, stream>>>(agg, W1, b0, xw, nTiles);
  if (remRow0 < N)
    gemm64_tail<true><<<((N - remRow0) * D + T - 1) / T, T, 0, stream>>>(agg, W1, b0, xw, remRow0, N);

  // 5) out = dinv^2 * xw + b1, then scatter edges into out
  self_msg<<<(N * (D / 4) + T - 1) / T, T, 0, stream>>>(xw, dinv, b1, out, N, 1);
  {
    long long threads = (long long)E * 16;
    int grid = (int)((threads + T - 1) / T);
    scatter_edges<<<grid, T, 0, stream>>>(edge_src, edge_dst, dinv, xw, out, E);
  }
}